// reduce_model_6854767804682
// MI455X (gfx1250) — compile-verified
//
#include <hip/hip_runtime.h>
#include <stdint.h>

#define HW 2048           // floats per row (64*32)
#define VEC4 (HW / 4)     // 512 float4 per row
#define GTHREADS 256      // 8 waves per block (wave32)

typedef float __attribute__((ext_vector_type(4))) v4f;

// ---------------- CSR build ----------------

__global__ __launch_bounds__(256) void zero_counts(unsigned* __restrict__ counts, int n) {
  int i = blockIdx.x * blockDim.x + threadIdx.x;
  if (i < n) counts[i] = 0u;
}

__global__ __launch_bounds__(256) void count_hist(const int* __restrict__ index,
                                                  unsigned* __restrict__ counts,
                                                  int n_src, int n_dst) {
  int i = blockIdx.x * blockDim.x + threadIdx.x;
  if (i < n_src) {
    int d = index[i];
    if ((unsigned)d < (unsigned)n_dst) atomicAdd(&counts[d], 1u);
  }
}

// Single-workgroup exclusive scan over up to 8192 bins (1024 thr x 8 bins).
__global__ __launch_bounds__(1024) void scan_offsets(const unsigned* __restrict__ counts,
                                                     unsigned* __restrict__ offsets,
                                                     unsigned* __restrict__ cursor,
                                                     int n_dst) {
  __shared__ unsigned part[1024];
  const int t = threadIdx.x;
  unsigned local[8];
  unsigned s = 0u;
#pragma unroll
  for (int j = 0; j < 8; ++j) {
    int idx = t * 8 + j;
    local[j] = (idx < n_dst) ? counts[idx] : 0u;
    s += local[j];
  }
  part[t] = s;
  __syncthreads();
  for (int off = 1; off < 1024; off <<= 1) {
    unsigned v = (t >= off) ? part[t - off] : 0u;
    __syncthreads();
    part[t] += v;
    __syncthreads();
  }
  unsigned b = part[t] - s;  // exclusive prefix for this thread's first bin
#pragma unroll
  for (int j = 0; j < 8; ++j) {
    int idx = t * 8 + j;
    if (idx < n_dst) { offsets[idx] = b; cursor[idx] = b; }
    b += local[j];
  }
  if (t == 1023) offsets[n_dst] = b;  // grand total
}

__global__ __launch_bounds__(256) void fill_srclist(const int* __restrict__ index,
                                                    unsigned* __restrict__ cursor,
                                                    unsigned* __restrict__ srclist,
                                                    int n_src, int n_dst) {
  int i = blockIdx.x * blockDim.x + threadIdx.x;
  if (i < n_src) {
    int d = index[i];
    if ((unsigned)d < (unsigned)n_dst) {
      unsigned pos = atomicAdd(&cursor[d], 1u);
      srclist[pos] = (unsigned)i;
    }
  }
}

// ---------------- gather-max over CSR ----------------

__device__ __forceinline__ v4f fmax4(v4f a, v4f b) {
  v4f r;
  r.x = __builtin_fmaxf(a.x, b.x);
  r.y = __builtin_fmaxf(a.y, b.y);
  r.z = __builtin_fmaxf(a.z, b.z);
  r.w = __builtin_fmaxf(a.w, b.w);
  return r;
}

__global__ __launch_bounds__(GTHREADS) void gather_max(
    const v4f* __restrict__ x, const v4f* __restrict__ t,
    const unsigned* __restrict__ offsets, const unsigned* __restrict__ srclist,
    v4f* __restrict__ out) {
  // Double-buffered LDS staging for t rows: 2 * 512 * 16B = 16 KB of 320 KB/WGP.
  __shared__ v4f buf[2][VEC4];

  const int d = blockIdx.x;
  const int tid = threadIdx.x;

  const v4f* xrow = x + (size_t)d * VEC4;
  v4f a0 = __builtin_nontemporal_load(&xrow[tid]);             // x read once: NT
  v4f a1 = __builtin_nontemporal_load(&xrow[tid + GTHREADS]);

  const unsigned beg = offsets[d];
  const unsigned end = offsets[d + 1];

  // Per-lane async global->LDS copy of this lane's two float4 slots of row `src`.
  // Each lane only ever reads LDS it wrote itself, so s_wait_asynccnt (per-wave,
  // in-order completion) is the only sync needed. t is a 256 MiB one-shot stream
  // (> 192 MB L2), so mark it non-temporal to keep x/out resident.
  auto issue = [&](unsigned k, int b) {
    unsigned src = srclist[k];
    unsigned long long g0 =
        (unsigned long long)(uintptr_t)(const void*)(t + (size_t)src * VEC4 + tid);
    unsigned long long g1 = g0 + (unsigned long long)GTHREADS * 16ull;
    unsigned l0 = (unsigned)(uintptr_t)(void*)&buf[b][tid];          // LDS byte addr
    unsigned l1 = l0 + (unsigned)GTHREADS * 16u;
    asm volatile("global_load_async_to_lds_b128 %0, %1, off th:TH_LOAD_NT\n\t"
                 "global_load_async_to_lds_b128 %2, %3, off th:TH_LOAD_NT"
                 :: "v"(l0), "v"(g0), "v"(l1), "v"(g1)
                 : "memory");
  };

  if (beg < end) {
    issue(beg, 0);
    for (unsigned k = beg; k < end; ++k) {
      const int b = (int)((k - beg) & 1u);
      if (k + 1u < end) {
        issue(k + 1u, b ^ 1);                       // prefetch next row (2 async ops)
        asm volatile("s_wait_asynccnt 2" ::: "memory");  // current row's 2 ops done
      } else {
        asm volatile("s_wait_asynccnt 0" ::: "memory");
      }
      v4f v0 = buf[b][tid];
      v4f v1 = buf[b][tid + GTHREADS];
      a0 = fmax4(a0, v0);
      a1 = fmax4(a1, v1);
    }
  }

  v4f* orow = out + (size_t)d * VEC4;
  __builtin_nontemporal_store(a0, &orow[tid]);     // out written once: NT
  __builtin_nontemporal_store(a1, &orow[tid + GTHREADS]);
}

// ---------------- launcher ----------------

extern "C" void kernel_launch(void* const* d_in, const int* in_sizes, int n_in,
                              void* d_out, int out_size, void* d_ws, size_t ws_size,
                              hipStream_t stream) {
  (void)n_in; (void)ws_size;
  const float* x = (const float*)d_in[0];
  const float* t = (const float*)d_in[1];
  const int* index = (const int*)d_in[2];
  float* out = (float*)d_out;

  const int n_src = in_sizes[2];       // 32768
  const int n_dst = out_size / HW;     // 8192

  char* ws = (char*)d_ws;
  size_t o = 0;
  unsigned* counts = (unsigned*)(ws + o);  o += (size_t)n_dst * 4;        o = (o + 15) & ~(size_t)15;
  unsigned* offsets = (unsigned*)(ws + o); o += (size_t)(n_dst + 1) * 4;  o = (o + 15) & ~(size_t)15;
  unsigned* cursor = (unsigned*)(ws + o);  o += (size_t)n_dst * 4;        o = (o + 15) & ~(size_t)15;
  unsigned* srclist = (unsigned*)(ws + o);

  zero_counts<<<(n_dst + 255) / 256, 256, 0, stream>>>(counts, n_dst);
  count_hist<<<(n_src + 255) / 256, 256, 0, stream>>>(index, counts, n_src, n_dst);
  scan_offsets<<<1, 1024, 0, stream>>>(counts, offsets, cursor, n_dst);
  fill_srclist<<<(n_src + 255) / 256, 256, 0, stream>>>(index, cursor, srclist, n_src, n_dst);
  gather_max<<<n_dst, GTHREADS, 0, stream>>>((const v4f*)x, (const v4f*)t, offsets, srclist,
                                             (v4f*)out);
}